// BinaryLogLoss_89678917140700
// MI455X (gfx1250) — compile-verified
//
#include <hip/hip_runtime.h>
#include <math.h>

typedef __attribute__((ext_vector_type(2))) float v2f;
typedef __attribute__((ext_vector_type(4))) float v4f;
typedef __attribute__((ext_vector_type(8))) float v8f;

#define TILE_M 16
#define DIM 128
#define LDS_STRIDE 132   // floats; 132 % 64 == 4 -> conflict-free fragment reads, 8B-aligned pairs

__device__ __forceinline__ float log_sigmoid(float x) {
    // numerically stable: min(x,0) - log1p(exp(-|x|))
    return fminf(x, 0.0f) - log1pf(expf(-fabsf(x)));
}

__global__ void __launch_bounds__(32)
nce_loss_wmma_kernel(const float* __restrict__ hidden,   // [N, D]
                     const int*   __restrict__ label,    // [N]
                     const int*   __restrict__ negs,     // [N, K]
                     const float* __restrict__ weight,   // [V, D]
                     float* __restrict__ out,            // scalar
                     int N, int K)
{
    __shared__ float shH[TILE_M * LDS_STRIDE];
    __shared__ float shE[2][TILE_M * LDS_STRIDE];

    const int lane = threadIdx.x;             // 0..31, one wave per block
    const int n0   = blockIdx.x * TILE_M;

    // per-lane gather-loop geometry: 16 rows x 8 float4 chunks, 4 full-wave iters
    // chunk = it*32 + lane; row = chunk>>3; c = chunk&7
    // ---- load hidden tile 16x32 -> LDS (coalesced float4)
    #pragma unroll
    for (int it = 0; it < 4; ++it) {
        int chunk = it * 32 + lane;
        int row   = chunk >> 3;
        int c     = chunk & 7;
        v4f v = *(const v4f*)(hidden + (size_t)(n0 + row) * DIM + c * 4);
        *(v4f*)(&shH[row * LDS_STRIDE + c * 4]) = v;
    }

    // ---- slot-0 (positive) indices live in lanes 0..15, broadcast via shfl
    int myIdx = 0;
    if (lane < TILE_M) myIdx = label[n0 + lane];

    v4f r[4];
    #pragma unroll
    for (int it = 0; it < 4; ++it) {
        int chunk = it * 32 + lane;
        int row   = chunk >> 3;
        int c     = chunk & 7;
        int rIdx  = __shfl(myIdx, row, 32);
        r[it] = *(const v4f*)(weight + (size_t)rIdx * DIM + c * 4);
    }
    #pragma unroll
    for (int it = 0; it < 4; ++it) {
        int chunk = it * 32 + lane;
        int row   = chunk >> 3;
        int c     = chunk & 7;
        *(v4f*)(&shE[0][row * LDS_STRIDE + c * 4]) = r[it];
    }
    __syncthreads();

    const float invK = 1.0f / (float)K;
    const int   mrow = lane & 15;             // A: M row / B: N col per lane
    const int   koff = (lane >> 4) << 1;      // 0 for lanes 0-15, 2 for lanes 16-31
    const bool  own  = (lane < 8) || (lane >= 24);
    const int   g    = lane & 7;

    float lsum = 0.0f;
    int   cur  = 0;

    for (int slot = 0; slot <= K; ++slot) {
        const bool has_next = (slot < K);

        // ---- software pipeline: issue next slot's gather NOW (in flight
        //      during this slot's WMMA work), store to the other buffer later
        if (has_next) {
            if (lane < TILE_M)
                myIdx = negs[(size_t)(n0 + lane) * K + slot];   // next slot = slot+1
            #pragma unroll
            for (int it = 0; it < 4; ++it) {
                int chunk = it * 32 + lane;
                int row   = chunk >> 3;
                int c     = chunk & 7;
                int rIdx  = __shfl(myIdx, row, 32);
                r[it] = *(const v4f*)(weight + (size_t)rIdx * DIM + c * 4);
            }
        }

        // ---- 16x16 score tile via V_WMMA_F32_16X16X4_F32, K-loop over 128
        const float* eb = &shE[cur][0];
        v8f acc0 = {};
        v8f acc1 = {};
        #pragma unroll
        for (int k0 = 0; k0 < DIM; k0 += 8) {
            v2f a0 = *(const v2f*)(&shH[mrow * LDS_STRIDE + k0 + koff]);
            v2f b0 = *(const v2f*)(&eb [mrow * LDS_STRIDE + k0 + koff]);
            acc0 = __builtin_amdgcn_wmma_f32_16x16x4_f32(
                false, a0, false, b0, (short)0, acc0, false, false);
            v2f a1 = *(const v2f*)(&shH[mrow * LDS_STRIDE + k0 + 4 + koff]);
            v2f b1 = *(const v2f*)(&eb [mrow * LDS_STRIDE + k0 + 4 + koff]);
            acc1 = __builtin_amdgcn_wmma_f32_16x16x4_f32(
                false, a1, false, b1, (short)0, acc1, false, false);
        }

        // ---- diagonal extraction: lanes 0-7 own D[m][m] m=lane,
        //      lanes 24-31 own m=lane-16; element index is lane&7 in both cases
        if (own) {
            float s = 0.0f;
            #pragma unroll
            for (int i = 0; i < 8; ++i) {
                float e = acc0[i] + acc1[i];
                s += (g == i) ? e : 0.0f;
            }
            lsum += (slot == 0) ? log_sigmoid(s) : invK * log_sigmoid(-s);
        }

        // ---- drain prefetched gather into the other buffer
        if (has_next) {
            float* ebn = &shE[cur ^ 1][0];
            #pragma unroll
            for (int it = 0; it < 4; ++it) {
                int chunk = it * 32 + lane;
                int row   = chunk >> 3;
                int c     = chunk & 7;
                *(v4f*)(&ebn[row * LDS_STRIDE + c * 4]) = r[it];
            }
            __syncthreads();
            cur ^= 1;
        }
    }

    // ---- wave32 reduction, then one atomic per tile
    #pragma unroll
    for (int off = 16; off > 0; off >>= 1)
        lsum += __shfl_xor(lsum, off, 32);
    if (lane == 0)
        atomicAdd(out, -lsum);                 // loss = -(pos_term + neg_term)
}

__global__ void zero_out_kernel(float* out, int n) {
    int i = blockIdx.x * blockDim.x + threadIdx.x;
    if (i < n) out[i] = 0.0f;
}

extern "C" void kernel_launch(void* const* d_in, const int* in_sizes, int n_in,
                              void* d_out, int out_size, void* d_ws, size_t ws_size,
                              hipStream_t stream) {
    const float* hidden = (const float*)d_in[0];   // [N, D] fp32
    const int*   label  = (const int*)  d_in[1];   // [N]
    const int*   negs   = (const int*)  d_in[2];   // [N, K]
    const float* weight = (const float*)d_in[3];   // [V, D] fp32
    float*       out    = (float*)d_out;

    const int N = in_sizes[1];
    const int K = in_sizes[2] / N;

    zero_out_kernel<<<1, 32, 0, stream>>>(out, out_size);

    const int tiles = N / TILE_M;                  // N=16384 -> 1024 tiles
    nce_loss_wmma_kernel<<<tiles, 32, 0, stream>>>(hidden, label, negs, weight,
                                                   out, N, K);
}